// ChannelAttention_1425929142543
// MI455X (gfx1250) — compile-verified
//
#include <hip/hip_runtime.h>
#include <cstddef>

// ---------------------------------------------------------------------------
// Channel attention (XCiT-style) for gfx1250, fp32 via V_WMMA_F32_16X16X4_F32
// B=8, N=4096, C=768, H=8, hd=96
// ---------------------------------------------------------------------------

#define B_ 8
#define N_ 4096
#define C_ 768
#define H_ 8
#define HD_ 96
#define C3_ 2304
#define EPS_ 1e-12f

typedef __attribute__((ext_vector_type(2))) float v2f;
typedef __attribute__((ext_vector_type(4))) float v4f;
typedef __attribute__((ext_vector_type(8))) float v8f;

__device__ __forceinline__ v8f wmma_f32(v2f a, v2f b, v8f c) {
    // D = A(16x4) * B(4x16) + C(16x16), fp32
    return __builtin_amdgcn_wmma_f32_16x16x4_f32(
        false, a, false, b, (short)0, c, false, false);
}

// CDNA5 async global->LDS staging (ASYNCcnt path), guarded for portability.
#if defined(__gfx1250__) && __has_builtin(__builtin_amdgcn_global_load_async_to_lds_b32)
#define CA_ASYNC_LDS 1
#else
#define CA_ASYNC_LDS 0
#endif

typedef __attribute__((address_space(1))) int* gi32_p;   // global int*
typedef __attribute__((address_space(3))) int* li32_p;   // LDS int*

__device__ __forceinline__ void ca_stage_b32(const float* gsrc, float* ldst) {
#if CA_ASYNC_LDS
    __builtin_amdgcn_global_load_async_to_lds_b32(
        (gi32_p)(void*)gsrc, (li32_p)(void*)ldst, 0, 0);
#else
    *ldst = *gsrc;
#endif
}

template <int N>
__device__ __forceinline__ void ca_wait_async() {
#if CA_ASYNC_LDS
#if __has_builtin(__builtin_amdgcn_s_wait_asynccnt)
    __builtin_amdgcn_s_wait_asynccnt(N);
#else
    asm volatile("s_wait_asynccnt %0" :: "i"(N) : "memory");
#endif
#endif
}

// ---------------------------------------------------------------------------
// Activation @ weight GEMM, async double-buffered LDS weight panels.
// Block = 256 threads = 8 waves; block tile = 128 rows x 64 cols.
// K loop in panels of 32; weight panel stored k-pair interleaved:
//   lds[(k>>1)*128 + col*2 + (k&1)]  ->  B fragment = one aligned ds_load_b64.
// Panel p+1 streams in via ASYNCcnt while panel p is consumed (8 asyncs per
// thread per panel; s_wait_asynccnt 8 drains the current panel only).
//   act : [M, K] row-major (K contiguous -> float2 A fragments)
//   wgt : [K, NCOLS] row-major
//   out : [M, NCOLS]
// ---------------------------------------------------------------------------
template <int K, int NCOLS, bool ADD_BIAS>
__device__ __forceinline__ void ca_gemm_block(const float* __restrict__ act,
                                              const float* __restrict__ wgt,
                                              const float* __restrict__ bias,
                                              float* __restrict__ out) {
    constexpr unsigned KP = 32;              // K rows per panel
    constexpr unsigned NP = K / KP;          // number of panels
    __shared__ float lds_w[2][KP / 2 * 128]; // 2 x 8KB, interleaved pairs

    const unsigned tid   = threadIdx.x;
    const unsigned wave  = tid >> 5;
    const unsigned lane  = tid & 31u;
    const unsigned l16   = lane & 15u;
    const unsigned khalf = lane >> 4;

    const unsigned nblkcol = NCOLS / 64;
    const unsigned blkrow  = blockIdx.x / nblkcol;
    const unsigned col0    = (blockIdx.x % nblkcol) * 64u;
    const unsigned row0    = blkrow * 128u + wave * 16u;

    const float* wbase = wgt + col0;

    // stage panel rows [kk, kk+32) of cols [col0, col0+64) into buf
    auto stage = [&](unsigned kk, float* buf) {
#pragma unroll
        for (unsigned i = 0; i < 8; ++i) {
            const unsigned e   = tid + 256u * i;   // 0..2047
            const unsigned k   = e >> 6;           // 0..31
            const unsigned col = e & 63u;          // 0..63
            ca_stage_b32(wbase + (size_t)(kk + k) * NCOLS + col,
                         &buf[(k >> 1) * 128u + col * 2u + (k & 1u)]);
        }
    };

    v8f acc[4] = {};
    const float* arow = act + (size_t)(row0 + l16) * K;

    stage(0, lds_w[0]);
    for (unsigned p = 0; p < NP; ++p) {
        if (p + 1 < NP) {
            stage((p + 1) * KP, lds_w[(p + 1) & 1u]);
            ca_wait_async<8>();   // drain current panel, next stays in flight
        } else {
            ca_wait_async<0>();
        }
        __syncthreads();

        const float* buf = lds_w[p & 1u];
        const unsigned kk = p * KP;
#pragma unroll
        for (unsigned k4 = 0; k4 < KP; k4 += 4) {
            const unsigned kloc = k4 + 2u * khalf;                 // even
            const v2f a = *(const v2f*)(arow + kk + kloc);
            const float* b0 = &buf[(kloc >> 1) * 128u + l16 * 2u];
#pragma unroll
            for (int t = 0; t < 4; ++t) {
                const v2f bf = *(const v2f*)(b0 + 32 * t);         // ds_load_b64
                acc[t] = wmma_f32(a, bf, acc[t]);
            }
        }
        __syncthreads();   // all waves done reading before this buf is re-staged
    }

    const unsigned m0 = 8u * khalf;
#pragma unroll
    for (int t = 0; t < 4; ++t) {
        const unsigned col = col0 + 16u * t + l16;
        const float bv = ADD_BIAS ? bias[col] : 0.0f;
#pragma unroll
        for (int j = 0; j < 8; ++j) {
            out[(size_t)(row0 + m0 + j) * NCOLS + col] = acc[t][j] + bv;
        }
    }
}

// Kernel 1: qkv[32768, 2304] = x[32768, 768] @ Wqkv[768, 2304]
// Grid: 256 row-blocks * 36 col-groups = 9216 blocks of 256.
__global__ __launch_bounds__(256)
void ca_qkv_gemm(const float* __restrict__ x, const float* __restrict__ w,
                 float* __restrict__ qkv) {
    ca_gemm_block<C_, C3_, false>(x, w, nullptr, qkv);
}

// Kernel 5: out[32768, 768] = out_tmp @ Wproj + bproj
// Grid: 256 row-blocks * 12 col-groups = 3072 blocks of 256.
__global__ __launch_bounds__(256)
void ca_proj(const float* __restrict__ otmp, const float* __restrict__ w,
             const float* __restrict__ bias, float* __restrict__ out) {
    ca_gemm_block<C_, C_, true>(otmp, w, bias, out);
}

// ---------------------------------------------------------------------------
// Kernel 2: per (b,h) unnormalized Gram G[96,96] = sum_n q[d,n]*k[e,n]
// q[d,n] = qkv[b, n, h*96 + d], k at +768. Contraction K = 4096.
// Grid: 64 bh * 36 tiles = 2304 waves = 288 blocks of 256.
// ---------------------------------------------------------------------------
__global__ __launch_bounds__(256)
void ca_gram(const float* __restrict__ qkv, float* __restrict__ gram) {
    const unsigned gw    = blockIdx.x * 8u + (threadIdx.x >> 5);
    const unsigned lane  = threadIdx.x & 31u;
    const unsigned l16   = lane & 15u;
    const unsigned khalf = lane >> 4;
    const unsigned bh    = gw / 36u;
    const unsigned t     = gw % 36u;
    const unsigned b     = bh >> 3;
    const unsigned h     = bh & 7u;
    const unsigned d0    = (t / 6u) * 16u;
    const unsigned e0    = (t % 6u) * 16u;

    const size_t bhoff = (size_t)b * N_ * C3_ + (size_t)h * HD_;
    v8f acc = {};

    for (unsigned nn = 0; nn < N_; nn += 4) {
        const unsigned n = nn + 2u * khalf;
        const float* r0 = qkv + bhoff + (size_t)n * C3_;
        const float* r1 = r0 + C3_;
        v2f a, bf;
        a.x  = r0[d0 + l16];          // q[d, n]
        a.y  = r1[d0 + l16];          // q[d, n+1]
        bf.x = r0[C_ + e0 + l16];     // k[e, n]
        bf.y = r1[C_ + e0 + l16];     // k[e, n+1]
        acc = wmma_f32(a, bf, acc);
    }

    float* g = gram + (size_t)bh * (HD_ * HD_);
    const unsigned m0 = 8u * khalf;
#pragma unroll
    for (int j = 0; j < 8; ++j) {
        g[(size_t)(d0 + m0 + j) * HD_ + e0 + l16] = acc[j];
    }
}

// ---------------------------------------------------------------------------
// Kernel 2b: row L2 norms of q and k over N (per b,h,channel).
// Grid: 64 blocks of 96 threads.
// ---------------------------------------------------------------------------
__global__ __launch_bounds__(96)
void ca_sumsq(const float* __restrict__ qkv, float* __restrict__ qn,
              float* __restrict__ kn) {
    const unsigned bh = blockIdx.x;
    const unsigned b  = bh >> 3;
    const unsigned h  = bh & 7u;
    const unsigned d  = threadIdx.x;

    const float* base = qkv + (size_t)b * N_ * C3_ + (size_t)h * HD_ + d;
    float sq = 0.f, sk = 0.f;
    for (unsigned n = 0; n < N_; ++n) {
        const float qv = base[(size_t)n * C3_];
        const float kv = base[(size_t)n * C3_ + C_];
        sq += qv * qv;
        sk += kv * kv;
    }
    qn[bh * HD_ + d] = sqrtf(sq);
    kn[bh * HD_ + d] = sqrtf(sk);
}

// ---------------------------------------------------------------------------
// Kernel 3: attn = softmax_e( G[d,e] * temp[h] / (max(|q_d|,eps)*max(|k_e|,eps)) )
// In-place over gram. Grid: 64 blocks of 128 threads (one row per thread).
// ---------------------------------------------------------------------------
__global__ __launch_bounds__(128)
void ca_softmax(float* __restrict__ gram, const float* __restrict__ qn,
                const float* __restrict__ kn, const float* __restrict__ temp) {
    __shared__ float kinv[HD_];
    const unsigned bh = blockIdx.x;
    const unsigned h  = bh & 7u;
    const unsigned d  = threadIdx.x;

    if (d < HD_) kinv[d] = 1.0f / fmaxf(kn[bh * HD_ + d], EPS_);
    __syncthreads();
    if (d >= HD_) return;

    float* row = gram + (size_t)bh * (HD_ * HD_) + (size_t)d * HD_;
    const float s = temp[h] / fmaxf(qn[bh * HD_ + d], EPS_);

    float mx = -3.402823466e38f;
    for (int e = 0; e < HD_; ++e) mx = fmaxf(mx, row[e] * s * kinv[e]);
    float sum = 0.f;
    for (int e = 0; e < HD_; ++e) sum += expf(row[e] * s * kinv[e] - mx);
    const float inv = 1.0f / sum;
    for (int e = 0; e < HD_; ++e) row[e] = expf(row[e] * s * kinv[e] - mx) * inv;
}

// ---------------------------------------------------------------------------
// Kernel 4: out_tmp[b, n, h*96+d] = sum_e attn[d,e] * v[e,n], per (b,h).
// A = attn (contig float2 over e), B = v (contig float2 over e). 16x64 tiles.
// Grid: 64 bh * 6 dtiles * 64 ngroups = 24576 waves = 3072 blocks of 256.
// ---------------------------------------------------------------------------
__global__ __launch_bounds__(256)
void ca_attnv(const float* __restrict__ qkv, const float* __restrict__ attn,
              float* __restrict__ otmp) {
    const unsigned gw    = blockIdx.x * 8u + (threadIdx.x >> 5);
    const unsigned lane  = threadIdx.x & 31u;
    const unsigned l16   = lane & 15u;
    const unsigned khalf = lane >> 4;
    const unsigned bh    = gw / 384u;
    const unsigned r     = gw % 384u;
    const unsigned dt    = r / 64u;
    const unsigned ng    = r % 64u;
    const unsigned b     = bh >> 3;
    const unsigned h     = bh & 7u;
    const unsigned d0    = dt * 16u;
    const unsigned n0    = ng * 64u;

    const float* arow = attn + (size_t)bh * (HD_ * HD_) + (size_t)(d0 + l16) * HD_;
    const size_t vbase = (size_t)b * N_ * C3_ + 2 * C_ + (size_t)h * HD_;

    v8f acc[4] = {};
    for (unsigned ee = 0; ee < HD_; ee += 4) {
        const unsigned e = ee + 2u * khalf;
        const v2f a = *(const v2f*)(arow + e);            // attn[d, e..e+1]
#pragma unroll
        for (int t = 0; t < 4; ++t) {
            const unsigned ncol = n0 + 16u * t + l16;
            const v2f bf = *(const v2f*)(qkv + vbase + (size_t)ncol * C3_ + e);
            acc[t] = wmma_f32(a, bf, acc[t]);
        }
    }

    const unsigned m0 = 8u * khalf;
#pragma unroll
    for (int t = 0; t < 4; ++t) {
        float* dp = otmp + (size_t)(b * N_ + n0 + 16u * t + l16) * C_
                         + (size_t)h * HD_ + d0 + m0;
#pragma unroll
        for (int j = 0; j < 8; ++j) dp[j] = acc[t][j];    // 8 contiguous floats
    }
}

// ---------------------------------------------------------------------------
extern "C" void kernel_launch(void* const* d_in, const int* in_sizes, int n_in,
                              void* d_out, int out_size, void* d_ws, size_t ws_size,
                              hipStream_t stream) {
    const float* x     = (const float*)d_in[0];  // [8, 4096, 768]
    const float* Wqkv  = (const float*)d_in[1];  // [768, 2304]
    const float* temp  = (const float*)d_in[2];  // [8, 1, 1]
    const float* Wproj = (const float*)d_in[3];  // [768, 768]
    const float* bproj = (const float*)d_in[4];  // [768]
    float* out = (float*)d_out;                  // [8, 4096, 768]

    // workspace layout (floats)
    float* ws    = (float*)d_ws;
    float* qkv   = ws;                                   // 8*4096*2304 = 75497472
    float* gram  = qkv  + (size_t)B_ * N_ * C3_;         // 64*96*96    =   589824
    float* qn    = gram + (size_t)B_ * H_ * HD_ * HD_;   // 64*96
    float* kn    = qn   + (size_t)B_ * H_ * HD_;         // 64*96
    float* otmp  = kn   + (size_t)B_ * H_ * HD_;         // 8*4096*768  = 25165824

    // 1) QKV GEMM: 256 row-blocks * 36 col-groups
    ca_qkv_gemm<<<9216, 256, 0, stream>>>(x, Wqkv, qkv);
    // 2) Gram matrices: 2304 waves
    ca_gram<<<288, 256, 0, stream>>>(qkv, gram);
    // 2b) row norms
    ca_sumsq<<<64, 96, 0, stream>>>(qkv, qn, kn);
    // 3) scale + softmax (in place over gram)
    ca_softmax<<<64, 128, 0, stream>>>(gram, qn, kn, temp);
    // 4) attn @ v : 24576 waves
    ca_attnv<<<3072, 256, 0, stream>>>(qkv, gram, otmp);
    // 5) projection + bias: 256 row-blocks * 12 col-groups
    ca_proj<<<3072, 256, 0, stream>>>(otmp, Wproj, bproj, out);
}